// ReadoutNeck_32006096290278
// MI455X (gfx1250) — compile-verified
//
#include <hip/hip_runtime.h>
#include <math.h>

typedef __attribute__((ext_vector_type(2))) float v2f;
typedef __attribute__((ext_vector_type(8))) float v8f;

#define NN 32
#define MM 2
#define CC 256
#define TV 2500            // T*V = 100*25
#define PP 8
#define TOK_PER_BLOCK 128  // 8 waves x 16 tokens
#define CHUNKS 20          // ceil(2500/128)
#define EPSF 1e-8f

#if __has_builtin(__builtin_amdgcn_wmma_f32_16x16x4_f32)
#define HAVE_WMMA_F32X4 1
#else
#warning "builtin __builtin_amdgcn_wmma_f32_16x16x4_f32 not found - using scalar fallback (wmma=0 expected)"
#define HAVE_WMMA_F32X4 0
#endif

__global__ void rn_zero_kernel(float* ws, int n) {
    int i = blockIdx.x * blockDim.x + threadIdx.x;
    if (i < n) ws[i] = 0.0f;
}

// grid = (CHUNKS, NN), block = 256 (8 waves of 32)
__global__ void __launch_bounds__(256)
rn_main_kernel(const float* __restrict__ x, const float* __restrict__ protos,
               float* __restrict__ g_sums, float* __restrict__ g_cnt) {
    // Swizzled proto tiles for WMMA A-operand: per K-chunk j, lane L holds
    // A[M=L%16][K = 4j + (L<16 ? {0,1} : {2,3})], rows 8..15 are zero padding.
    __shared__ __align__(16) float ldsA[64 * 64];         // 16 KB
    __shared__ float lds_pn[PP];
    __shared__ __align__(16) float lds_sums[PP * 257];    // padded rows -> bank = (p+c)%64
    __shared__ float lds_cnt[PP];

    const int tid   = threadIdx.x;
    const int chunk = blockIdx.x;
    const int n     = blockIdx.y;

    // ---- stage swizzled protos into LDS ----
    for (int idx = tid; idx < 64 * 64; idx += 256) {
        int j     = idx >> 6;
        int rem   = idx & 63;
        int L     = rem >> 1;
        int which = rem & 1;
        int m     = L & 15;
        int k     = 4 * j + ((L < 16) ? which : (2 + which));
        ldsA[idx] = (m < PP) ? protos[m * CC + k] : 0.0f;
    }
    // ---- prototype norms ----
    if (tid < PP) {
        float s = 0.0f;
        for (int c = 0; c < CC; ++c) { float pv = protos[tid * CC + c]; s += pv * pv; }
        lds_pn[tid] = sqrtf(s);
    }
    // ---- zero block-local partials ----
    for (int idx = tid; idx < PP * 257; idx += 256) lds_sums[idx] = 0.0f;
    if (tid < PP) lds_cnt[tid] = 0.0f;
    __syncthreads();

    const int lane  = tid & 31;
    const int wave  = tid >> 5;
    const int tloc  = lane & 15;                       // token slot within wave tile
    const int tv    = chunk * TOK_PER_BLOCK + wave * 16 + tloc;
    const bool valid = (tv < TV);
    const int tvc   = valid ? tv : (TV - 1);           // clamp for addressing
    const int khalf = (lane >> 4) << 1;                // 0 (lanes 0-15) or 2 (lanes 16-31)

    const float* xm0 = x + (size_t)n * MM * CC * TV;   // person 0
    const float* xm1 = xm0 + (size_t)CC * TV;          // person 1

    // ---- pass 1: dots with prototypes via WMMA f32 16x16x4, plus token norm ----
    v8f acc = (v8f){0.f, 0.f, 0.f, 0.f, 0.f, 0.f, 0.f, 0.f};
#if !HAVE_WMMA_F32X4
    float fdot[PP];
#pragma unroll
    for (int p = 0; p < PP; ++p) fdot[p] = 0.0f;
#endif
    float hh = 0.0f;

    for (int j = 0; j < 64; ++j) {
        int c0 = 4 * j + khalf;
        size_t o0 = (size_t)c0 * TV + tvc;
        size_t o1 = o0 + TV;
        float b0 = 0.5f * (xm0[o0] + xm1[o0]);         // h[token][c0]
        float b1 = 0.5f * (xm0[o1] + xm1[o1]);         // h[token][c0+1]
        hh += b0 * b0 + b1 * b1;
#if HAVE_WMMA_F32X4
        v2f a = *(const v2f*)&ldsA[(j << 6) + (lane << 1)];
        v2f b; b.x = b0; b.y = b1;
        acc = __builtin_amdgcn_wmma_f32_16x16x4_f32(false, a, false, b,
                                                    (short)0, acc, false, false);
#else
#pragma unroll
        for (int p = 0; p < PP; ++p)
            fdot[p] += b0 * protos[p * CC + c0] + b1 * protos[p * CC + c0 + 1];
#endif
    }

    // combine half-lane partials (token t lives on lanes t and t+16 with disjoint c)
    hh += __shfl_xor(hh, 16, 32);
    float hn = sqrtf(hh);

    float dots[PP];
#if HAVE_WMMA_F32X4
    // lanes 0-15 hold D[M=p][N=token] for p=0..7 in acc[0..7]
#pragma unroll
    for (int p = 0; p < PP; ++p) dots[p] = acc[p];
#else
#pragma unroll
    for (int p = 0; p < PP; ++p) dots[p] = fdot[p] + __shfl_xor(fdot[p], 16, 32);
#endif

    // argmax of cosine similarity (strict > => first max, matches jnp.argmax)
    float best = -3.0e38f;
    int bi = 0;
#pragma unroll
    for (int p = 0; p < PP; ++p) {
        float sim = dots[p] / fmaxf(hn * lds_pn[p], EPSF);
        if (sim > best) { best = sim; bi = p; }
    }
    // broadcast each token's assignment to both half-lanes
    int assign = __shfl(bi, tloc, 32);

    if (lane < 16 && valid)
        __hip_atomic_fetch_add(&lds_cnt[assign], 1.0f,
                               __ATOMIC_RELAXED, __HIP_MEMORY_SCOPE_WORKGROUP);

    // ---- pass 2: re-sweep x (L2-resident) and accumulate h into segment sums ----
    for (int i = 0; i < 128; ++i) {
        int c = 2 * i + (lane >> 4);
        size_t o = (size_t)c * TV + tvc;
        float h = 0.5f * (xm0[o] + xm1[o]);
        if (valid)
            __hip_atomic_fetch_add(&lds_sums[assign * 257 + c], h,
                                   __ATOMIC_RELAXED, __HIP_MEMORY_SCOPE_WORKGROUP);
    }
    __syncthreads();

    // ---- flush block partials to global ----
    float* gs = g_sums + (size_t)n * PP * CC;
#pragma unroll
    for (int p = 0; p < PP; ++p)
        __hip_atomic_fetch_add(&gs[p * CC + tid], lds_sums[p * 257 + tid],
                               __ATOMIC_RELAXED, __HIP_MEMORY_SCOPE_AGENT);
    if (tid < PP)
        __hip_atomic_fetch_add(&g_cnt[n * PP + tid], lds_cnt[tid],
                               __ATOMIC_RELAXED, __HIP_MEMORY_SCOPE_AGENT);
}

// grid = NN, block = CC
__global__ void rn_final_kernel(const float* __restrict__ g_sums,
                                const float* __restrict__ g_cnt,
                                float* __restrict__ out) {
    int n = blockIdx.x;
    int c = threadIdx.x;
    float s = 0.0f;
#pragma unroll
    for (int p = 0; p < PP; ++p) {
        float cnt = g_cnt[n * PP + p];
        s += g_sums[((size_t)n * PP + p) * CC + c] / fmaxf(cnt, 1.0f);
    }
    out[n * CC + c] = s * 0.125f;
}

extern "C" void kernel_launch(void* const* d_in, const int* in_sizes, int n_in,
                              void* d_out, int out_size, void* d_ws, size_t ws_size,
                              hipStream_t stream) {
    const float* x      = (const float*)d_in[0];   // (32,2,256,100,25) f32
    const float* protos = (const float*)d_in[1];   // (8,256) f32

    float* g_sums = (float*)d_ws;                  // NN*PP*CC floats
    float* g_cnt  = g_sums + NN * PP * CC;         // NN*PP floats

    int zn = NN * PP * CC + NN * PP;
    rn_zero_kernel<<<(zn + 255) / 256, 256, 0, stream>>>((float*)d_ws, zn);

    dim3 grid(CHUNKS, NN);
    rn_main_kernel<<<grid, 256, 0, stream>>>(x, protos, g_sums, g_cnt);

    rn_final_kernel<<<NN, CC, 0, stream>>>(g_sums, g_cnt, (float*)d_out);
}